// AFA_90795608637594
// MI455X (gfx1250) — compile-verified
//
#include <hip/hip_runtime.h>
#include <math.h>

// ---------------------------------------------------------------------------
// Problem constants (match reference)
// ---------------------------------------------------------------------------
#define N_TOT   1024      // total atoms
#define DIM     128
#define NMOL    64        // molecules
#define NATOM   16        // atoms per molecule
#define NORM_EPS 1e-12f
#define OFFBLK  1008.0f   // 1024 - 16 off-block entries per bond row

typedef float v2f __attribute__((ext_vector_type(2)));
typedef float v8f __attribute__((ext_vector_type(8)));

// D = A(16x4 f32) * B(4x16 f32) + C(16x16 f32), wave32 WMMA
__device__ __forceinline__ v8f wmma4(v2f a, v2f b, v8f c) {
    return __builtin_amdgcn_wmma_f32_16x16x4_f32(
        /*neg_a=*/false, a, /*neg_b=*/false, b,
        /*c_mod=*/(short)0, c, /*reuse_a=*/false, /*reuse_b=*/false);
}

// ---------------------------------------------------------------------------
// 1) mps_raw[r, :] = E_fp[fingerprints[r], :]
// ---------------------------------------------------------------------------
__global__ void k_gather_rows(const int* __restrict__ idx,
                              const float* __restrict__ E,
                              float* __restrict__ out) {
    int r = blockIdx.x;          // 1024
    int c = threadIdx.x;         // 128
    out[r * DIM + c] = E[(size_t)idx[r] * DIM + c];
}

// ---------------------------------------------------------------------------
// 2) C[M,128] = act(A[M,128] @ W[128,128] + bias)   (WMMA f32 16x16x4)
//    one wave per 16x16 output tile; wave id -> (tm, tn)
// ---------------------------------------------------------------------------
template <bool RELU>
__global__ void k_gemm_wmma(const float* __restrict__ A,
                            const float* __restrict__ W,
                            const float* __restrict__ bias,
                            float* __restrict__ C, int M) {
    int wave = (int)((blockIdx.x * blockDim.x + threadIdx.x) >> 5);
    int lane = (int)(threadIdx.x & 31);
    int tm = wave >> 3;              // DIM/16 == 8 N-tiles
    int tn = wave & 7;
    if (tm * 16 >= M) return;        // wave-uniform guard

    int half = lane >> 4;            // 0: K pair {0,1}; 1: K pair {2,3}
    int l    = lane & 15;
    v8f acc = {};
    const float* Arow = A + (size_t)(tm * 16 + l) * DIM;
    for (int k = 0; k < DIM; k += 4) {
        int ka = k + 2 * half;
        v2f a, b;
        a.x = Arow[ka];
        a.y = Arow[ka + 1];
        b.x = W[(size_t)ka       * DIM + tn * 16 + l];
        b.y = W[(size_t)(ka + 1) * DIM + tn * 16 + l];
        acc = wmma4(a, b, acc);
    }
    float bb = bias[tn * 16 + l];
    #pragma unroll
    for (int r = 0; r < 8; ++r) {
        int row = tm * 16 + r + half * 8;
        float v = acc[r] + bb;
        if (RELU) v = v > 0.0f ? v : 0.0f;
        C[(size_t)row * DIM + tn * 16 + l] = v;
    }
}

// ---------------------------------------------------------------------------
// 3) mps_mp = mps_raw + Adj_block(16x16) @ contri_block(16x128)   (WMMA, K=16)
//    one wave per (molecule, n-tile); 64*8 = 512 waves
// ---------------------------------------------------------------------------
__global__ void k_msgpass_wmma(const float* __restrict__ Adj,
                               const float* __restrict__ contri,
                               const float* __restrict__ mps_raw,
                               float* __restrict__ mps_out) {
    int wave = (int)((blockIdx.x * blockDim.x + threadIdx.x) >> 5);
    int lane = (int)(threadIdx.x & 31);
    int mol  = wave >> 3;
    int tn   = wave & 7;
    int m0 = mol * NATOM;
    int n0 = tn * 16;
    int half = lane >> 4;
    int l    = lane & 15;
    v8f acc = {};
    #pragma unroll
    for (int k = 0; k < NATOM; k += 4) {
        int ka = k + 2 * half;
        v2f a, b;
        a.x = Adj[(size_t)(m0 + l) * N_TOT + m0 + ka];
        a.y = Adj[(size_t)(m0 + l) * N_TOT + m0 + ka + 1];
        b.x = contri[(size_t)(m0 + ka)     * DIM + n0 + l];
        b.y = contri[(size_t)(m0 + ka + 1) * DIM + n0 + l];
        acc = wmma4(a, b, acc);
    }
    #pragma unroll
    for (int r = 0; r < 8; ++r) {
        int row = m0 + r + half * 8;
        int col = n0 + l;
        mps_out[(size_t)row * DIM + col] =
            mps_raw[(size_t)row * DIM + col] + acc[r];
    }
}

// ---------------------------------------------------------------------------
// 4) row-wise L2 normalize X[1024,128] in place (F.normalize semantics)
// ---------------------------------------------------------------------------
__global__ void k_rownorm(float* __restrict__ X) {
    int r = blockIdx.x;              // 1024
    int c = threadIdx.x;             // 128 threads = 4 waves
    float v = X[(size_t)r * DIM + c];
    float s = v * v;
    #pragma unroll
    for (int o = 16; o; o >>= 1) s += __shfl_xor(s, o);
    __shared__ float red[4];
    if ((threadIdx.x & 31) == 0) red[threadIdx.x >> 5] = s;
    __syncthreads();
    float tot = red[0] + red[1] + red[2] + red[3];
    float n = fmaxf(sqrtf(tot), NORM_EPS);
    X[(size_t)r * DIM + c] = v / n;
}

// ---------------------------------------------------------------------------
// 5) mpo column norms + mpshat = mps_fin / norm_mpo
//    norm_mpo[a,c]^2 = sum_{b in mol(a)} E_bond[idx[a,b],c]^2 + 1008*E0[c]^2
// ---------------------------------------------------------------------------
__global__ void k_mpshat(const int* __restrict__ bond_index,
                         const float* __restrict__ E_bond,
                         const float* __restrict__ mps_fin,
                         float* __restrict__ mpshat) {
    int a = blockIdx.x;              // 1024
    int c = threadIdx.x;             // 128
    int m0 = a & ~(NATOM - 1);
    __shared__ int idx[NATOM];
    if (c < NATOM) idx[c] = bond_index[(size_t)a * N_TOT + m0 + c];
    __syncthreads();
    float e0 = E_bond[c];
    float s = OFFBLK * e0 * e0;
    #pragma unroll
    for (int j = 0; j < NATOM; ++j) {
        float e = E_bond[(size_t)idx[j] * DIM + c];
        s += e * e;
    }
    float n = fmaxf(sqrtf(s), NORM_EPS);
    mpshat[(size_t)a * DIM + c] = mps_fin[(size_t)a * DIM + c] / n;
}

// ---------------------------------------------------------------------------
// 6) Sm[m,c] = sum_{a in m} mpshat[a,c];  7) Sall[c] = sum_m Sm[m,c]
// ---------------------------------------------------------------------------
__global__ void k_Sm(const float* __restrict__ mpshat, float* __restrict__ Sm) {
    int m = blockIdx.x, c = threadIdx.x;
    float s = 0.f;
    #pragma unroll
    for (int i = 0; i < NATOM; ++i) s += mpshat[(size_t)(m * NATOM + i) * DIM + c];
    Sm[(size_t)m * DIM + c] = s;
}
__global__ void k_Sall(const float* __restrict__ Sm, float* __restrict__ Sall) {
    int c = threadIdx.x;
    float s = 0.f;
    for (int m = 0; m < NMOL; ++m) s += Sm[(size_t)m * DIM + c];
    Sall[c] = s;
}

// ---------------------------------------------------------------------------
// 8) tn[m,c] = sum_{a,b in m} mpshat[a,c]*E_bond[idx[a,b],c]*mps_fin[b,c]
//            + (sum_b mps_fin[b,c]) * (Sall[c]-Sm[m,c]) * E_bond[0,c]
// ---------------------------------------------------------------------------
__global__ void k_tn(const int* __restrict__ bond_index,
                     const float* __restrict__ E_bond,
                     const float* __restrict__ mps_fin,
                     const float* __restrict__ mpshat,
                     const float* __restrict__ Sm,
                     const float* __restrict__ Sall,
                     float* __restrict__ tn) {
    int m = blockIdx.x;              // 64
    int c = threadIdx.x;             // 128
    int B0 = m * NATOM;
    __shared__ int   idx[NATOM][NATOM];
    __shared__ float shat[NATOM][DIM];
    __shared__ float sfin[NATOM][DIM];
    for (int t = c; t < NATOM * NATOM; t += DIM)
        idx[t >> 4][t & 15] =
            bond_index[(size_t)(B0 + (t >> 4)) * N_TOT + B0 + (t & 15)];
    #pragma unroll
    for (int i = 0; i < NATOM; ++i) {
        shat[i][c] = mpshat[(size_t)(B0 + i) * DIM + c];
        sfin[i][c] = mps_fin[(size_t)(B0 + i) * DIM + c];
    }
    __syncthreads();
    float F = 0.f;
    #pragma unroll
    for (int b = 0; b < NATOM; ++b) F += sfin[b][c];
    float acc = 0.f;
    for (int a = 0; a < NATOM; ++a) {
        float inner = 0.f;
        #pragma unroll
        for (int b = 0; b < NATOM; ++b)
            inner += E_bond[(size_t)idx[a][b] * DIM + c] * sfin[b][c];
        acc += shat[a][c] * inner;
    }
    float e0 = E_bond[c];
    acc += F * (Sall[c] - Sm[(size_t)m * DIM + c]) * e0;
    tn[(size_t)m * DIM + c] = acc;
}

// ---------------------------------------------------------------------------
// 9) out[m] = h1[m,:] . W_prop[:,0] + b_prop
// ---------------------------------------------------------------------------
__global__ void k_prop(const float* __restrict__ h1,
                       const float* __restrict__ Wp,
                       const float* __restrict__ bp,
                       float* __restrict__ out) {
    int m = blockIdx.x;              // 64
    int c = threadIdx.x;             // 128 threads = 4 waves
    float v = h1[(size_t)m * DIM + c] * Wp[c];
    #pragma unroll
    for (int o = 16; o; o >>= 1) v += __shfl_xor(v, o);
    __shared__ float red[4];
    if ((threadIdx.x & 31) == 0) red[threadIdx.x >> 5] = v;
    __syncthreads();
    if (threadIdx.x == 0)
        out[m] = red[0] + red[1] + red[2] + red[3] + bp[0];
}

// ---------------------------------------------------------------------------
// launch
// ---------------------------------------------------------------------------
extern "C" void kernel_launch(void* const* d_in, const int* in_sizes, int n_in,
                              void* d_out, int out_size, void* d_ws, size_t ws_size,
                              hipStream_t stream) {
    const int*   fingerprints = (const int*)  d_in[0];   // [1024]
    const float* adjacency    = (const float*)d_in[1];   // [1024,1024]
    const int*   bond_index   = (const int*)  d_in[2];   // [1024,1024]
    const float* E_fp         = (const float*)d_in[3];   // [10000,128]
    const float* E_bond       = (const float*)d_in[4];   // [10000,128]
    const float* W_fp         = (const float*)d_in[5];   // [128,128]
    const float* b_fp         = (const float*)d_in[6];   // [128]
    const float* W_out0       = (const float*)d_in[7];
    const float* b_out0       = (const float*)d_in[8];
    const float* W_out1       = (const float*)d_in[9];
    const float* b_out1       = (const float*)d_in[10];
    const float* W_prop       = (const float*)d_in[11];  // [128,1]
    const float* b_prop       = (const float*)d_in[12];  // [1]
    float* out = (float*)d_out;                          // [64]

    float* ws      = (float*)d_ws;
    float* mps_raw = ws;                      // 1024*128
    float* contri  = mps_raw + N_TOT * DIM;   // 1024*128
    float* mps_fin = contri  + N_TOT * DIM;   // 1024*128 (msg-passed, then normed)
    float* mpshat  = mps_fin + N_TOT * DIM;   // 1024*128
    float* Sm      = mpshat  + N_TOT * DIM;   // 64*128
    float* Sall    = Sm      + NMOL * DIM;    // 128
    float* tn      = Sall    + DIM;           // 64*128
    float* h0      = tn      + NMOL * DIM;    // 64*128
    float* h1      = h0      + NMOL * DIM;    // 64*128

    // 1) atom embedding gather
    k_gather_rows<<<N_TOT, DIM, 0, stream>>>(fingerprints, E_fp, mps_raw);

    // 2) contri = relu(mps @ W_fp + b_fp)   [1024x128x128, WMMA]
    //    tiles: 64*8 waves = 16384 threads = 64 blocks of 256
    k_gemm_wmma<true><<<64, 256, 0, stream>>>(mps_raw, W_fp, b_fp, contri, N_TOT);

    // 3) block-diagonal message passing (WMMA, K=16): 64*8 waves = 64 blocks
    k_msgpass_wmma<<<64, 256, 0, stream>>>(adjacency, contri, mps_raw, mps_fin);

    // 4) row L2 normalize
    k_rownorm<<<N_TOT, DIM, 0, stream>>>(mps_fin);

    // 5) mpo norms + mpshat
    k_mpshat<<<N_TOT, DIM, 0, stream>>>(bond_index, E_bond, mps_fin, mpshat);

    // 6/7) per-molecule and global column sums
    k_Sm  <<<NMOL, DIM, 0, stream>>>(mpshat, Sm);
    k_Sall<<<1,    DIM, 0, stream>>>(Sm, Sall);

    // 8) tn contraction
    k_tn<<<NMOL, DIM, 0, stream>>>(bond_index, E_bond, mps_fin, mpshat, Sm, Sall, tn);

    // 9/10) MLP head (WMMA): M=64 -> 4*8 = 32 waves = 4 blocks of 256
    k_gemm_wmma<true><<<4, 256, 0, stream>>>(tn, W_out0, b_out0, h0, NMOL);
    k_gemm_wmma<true><<<4, 256, 0, stream>>>(h0, W_out1, b_out1, h1, NMOL);

    // 11) final projection to [64,1]
    k_prop<<<NMOL, DIM, 0, stream>>>(h1, W_prop, b_prop, out);
}